// align_loss_68745246539824
// MI455X (gfx1250) — compile-verified
//
#include <hip/hip_runtime.h>

typedef __attribute__((ext_vector_type(2))) float v2f;
typedef __attribute__((ext_vector_type(8))) float v8f;

#define NROW 8192
#define DD   128
#define ST   132   // LDS row stride in floats: 132 % 64 = 4 -> conflict-free b64 frag loads

// ---------------------------------------------------------------------------
// fc128: Y[128 rows] = X @ W^T + b   (D = 128, one 128-row band per block)
// 256 threads = 8 waves; wave w handles rows [w*16, w*16+16) x all 128 cols.
// ---------------------------------------------------------------------------
__global__ __launch_bounds__(256) void fc128(const float* __restrict__ X,
                                             const float* __restrict__ W,
                                             const float* __restrict__ bias,
                                             float* __restrict__ Y) {
    __shared__ float sX[128 * ST];
    __shared__ float sW[128 * ST];
    const int tid = threadIdx.x;
    const int rowbase = blockIdx.x * 128;

    // cooperative load: 128x128 floats each, float4-coalesced
    for (int i = tid; i < 128 * 32; i += 256) {
        const int r  = i >> 5;
        const int c4 = (i & 31) << 2;
        *(float4*)&sX[r * ST + c4] = *(const float4*)(X + (size_t)(rowbase + r) * DD + c4);
        *(float4*)&sW[r * ST + c4] = *(const float4*)(W + (size_t)r * DD + c4);
    }
    __syncthreads();

    const int wave  = tid >> 5;
    const int lane  = tid & 31;
    const int lrow  = lane & 15;
    const int khalf = lane >> 4;

    v8f acc[8] = {};
    for (int k0 = 0; k0 < DD; k0 += 4) {
        const v2f a = *(const v2f*)&sX[(wave * 16 + lrow) * ST + k0 + 2 * khalf];
#pragma unroll
        for (int jt = 0; jt < 8; ++jt) {
            const v2f b = *(const v2f*)&sW[(jt * 16 + lrow) * ST + k0 + 2 * khalf];
            acc[jt] = __builtin_amdgcn_wmma_f32_16x16x4_f32(
                false, a, false, b, (short)0, acc[jt], false, false);
        }
    }

#pragma unroll
    for (int jt = 0; jt < 8; ++jt) {
        const int   col = jt * 16 + lrow;
        const float bv  = bias[col];
#pragma unroll
        for (int v = 0; v < 8; ++v) {
            const int row = rowbase + wave * 16 + v + 8 * khalf;
            Y[(size_t)row * DD + col] = acc[jt][v] + bv;
        }
    }
}

// ---------------------------------------------------------------------------
// mmd128: one 128x128 tile of the pair matrix per block.
// blockIdx.z: 0 -> (xt,xt,+1)  1 -> (ti,ti,+1)  2 -> (xt,ti,-2)
// partial[z*4096 + bi*64 + bj] = weight * sum_tile exp(-d2/2)
// ---------------------------------------------------------------------------
__global__ __launch_bounds__(256) void mmd128(const float* __restrict__ xt,
                                              const float* __restrict__ ti,
                                              float* __restrict__ partial) {
    __shared__ float sA[128 * ST];
    __shared__ float sB[128 * ST];
    __shared__ float a2s[128];
    __shared__ float b2s[128];
    __shared__ float red[8];

    const int tid = threadIdx.x;
    const int bi  = blockIdx.x;
    const int bj  = blockIdx.y;
    const int z   = blockIdx.z;

    const float* A = (z == 1) ? ti : xt;
    const float* B = (z == 0) ? xt : ti;
    const float  weight = (z == 2) ? -2.0f : 1.0f;

    const size_t abase = (size_t)bi * 128 * DD;
    const size_t bbase = (size_t)bj * 128 * DD;
    for (int i = tid; i < 128 * 32; i += 256) {
        const int r  = i >> 5;
        const int c4 = (i & 31) << 2;
        *(float4*)&sA[r * ST + c4] = *(const float4*)(A + abase + (size_t)r * DD + c4);
        *(float4*)&sB[r * ST + c4] = *(const float4*)(B + bbase + (size_t)r * DD + c4);
    }
    __syncthreads();

    // squared row norms from LDS (threads 0-127: A rows, 128-255: B rows)
    {
        const float* src = (tid < 128) ? &sA[(tid)*ST] : &sB[(tid - 128) * ST];
        float s = 0.f;
        for (int k = 0; k < DD; k += 4) {
            const float4 q = *(const float4*)&src[k];
            s += q.x * q.x + q.y * q.y + q.z * q.z + q.w * q.w;
        }
        if (tid < 128) a2s[tid] = s; else b2s[tid - 128] = s;
    }
    __syncthreads();

    const int wave  = tid >> 5;     // i-band 0..7
    const int lane  = tid & 31;
    const int lrow  = lane & 15;
    const int khalf = lane >> 4;

    v8f acc[8] = {};
    for (int k0 = 0; k0 < DD; k0 += 4) {
        const v2f a = *(const v2f*)&sA[(wave * 16 + lrow) * ST + k0 + 2 * khalf];
#pragma unroll
        for (int jt = 0; jt < 8; ++jt) {
            const v2f b = *(const v2f*)&sB[(jt * 16 + lrow) * ST + k0 + 2 * khalf];
            acc[jt] = __builtin_amdgcn_wmma_f32_16x16x4_f32(
                false, a, false, b, (short)0, acc[jt], false, false);
        }
    }

    // epilogue: d2 = |a|^2 + |b|^2 - 2 a.b ; sum exp(-d2/2)
    float aa[8];
#pragma unroll
    for (int v = 0; v < 8; ++v) aa[v] = a2s[wave * 16 + v + 8 * khalf];

    const float K = -0.72134752044448170368f;  // -1/(2*ln2)
    float local = 0.f;
#pragma unroll
    for (int jt = 0; jt < 8; ++jt) {
        const float bb = b2s[jt * 16 + lrow];
#pragma unroll
        for (int v = 0; v < 8; ++v) {
            float d2 = aa[v] + bb - 2.0f * acc[jt][v];
            d2 = fmaxf(d2, 0.0f);
            local += __builtin_exp2f(d2 * K);
        }
    }

    // deterministic reduction: shfl tree within wave32, then LDS across waves
#pragma unroll
    for (int off = 16; off > 0; off >>= 1) local += __shfl_xor(local, off, 32);
    if (lane == 0) red[wave] = local;
    __syncthreads();
    if (tid == 0) {
        float s = 0.f;
#pragma unroll
        for (int w = 0; w < 8; ++w) s += red[w];
        partial[((size_t)z * 64 + bi) * 64 + bj] = s * weight;
    }
}

// ---------------------------------------------------------------------------
// final fixed-order reduction of 3*64*64 partials -> scalar
// ---------------------------------------------------------------------------
__global__ __launch_bounds__(256) void final_reduce(const float* __restrict__ partial,
                                                    int n, float* __restrict__ out,
                                                    float scale) {
    __shared__ float red[8];
    const int tid = threadIdx.x;
    float s = 0.f;
    for (int i = tid; i < n; i += 256) s += partial[i];
#pragma unroll
    for (int off = 16; off > 0; off >>= 1) s += __shfl_xor(s, off, 32);
    if ((tid & 31) == 0) red[tid >> 5] = s;
    __syncthreads();
    if (tid == 0) {
        float t = 0.f;
#pragma unroll
        for (int w = 0; w < 8; ++w) t += red[w];
        out[0] = t * scale;
    }
}

extern "C" void kernel_launch(void* const* d_in, const int* in_sizes, int n_in,
                              void* d_out, int out_size, void* d_ws, size_t ws_size,
                              hipStream_t stream) {
    const float* x_text = (const float*)d_in[0];
    const float* x_time = (const float*)d_in[1];
    const float* W1     = (const float*)d_in[2];
    const float* b1     = (const float*)d_in[3];
    const float* W2     = (const float*)d_in[4];
    const float* b2     = (const float*)d_in[5];

    float* xt      = (float*)d_ws;                 // 8192*128 floats
    float* tiq     = xt + (size_t)NROW * DD;       // 8192*128 floats
    float* partial = tiq + (size_t)NROW * DD;      // 3*64*64 floats

    fc128<<<NROW / 128, 256, 0, stream>>>(x_text, W1, b1, xt);
    fc128<<<NROW / 128, 256, 0, stream>>>(x_time, W2, b2, tiq);

    dim3 grid(64, 64, 3);
    mmd128<<<grid, 256, 0, stream>>>(xt, tiq, partial);

    final_reduce<<<1, 256, 0, stream>>>(partial, 3 * 64 * 64, (float*)d_out,
                                        1.0f / ((float)NROW * (float)NROW));
}